// EfficientAdditiveAttnetion_51247549776064
// MI455X (gfx1250) — compile-verified
//
#include <hip/hip_runtime.h>

// Problem constants (from reference setup_inputs)
#define B_   32
#define N_   1024
#define M_   1024
#define K2_  512
#define D_   768
#define SCALE_ 0.0625f      // 256^-0.5
#define EPS_   1e-12f

typedef __attribute__((ext_vector_type(2))) float v2f;
typedef __attribute__((ext_vector_type(8))) float v8f;

// ---------------------------------------------------------------------------
// K1: qw[b*N+n] = SCALE * dot(x[b,n,:], w_g)    (one wave32 per row)
// ---------------------------------------------------------------------------
__global__ void k_qw(const float* __restrict__ x, const float* __restrict__ wg,
                     float* __restrict__ qw) {
    int wave = threadIdx.x >> 5;
    int lane = threadIdx.x & 31;
    int row  = blockIdx.x * 8 + wave;           // 0 .. B*N-1
    const float* xr = x + (size_t)row * D_;
    float acc = 0.f;
    #pragma unroll 4
    for (int i = lane; i < D_; i += 32) acc += xr[i] * wg[i];
    #pragma unroll
    for (int off = 16; off > 0; off >>= 1) acc += __shfl_down(acc, off, 32);
    if (lane == 0) qw[row] = acc * SCALE_;
}

// ---------------------------------------------------------------------------
// K2: invn[b] = 1 / max(sqrt(sum_n qw^2), EPS)
// ---------------------------------------------------------------------------
__global__ void k_norm(const float* __restrict__ qw, float* __restrict__ invn) {
    __shared__ float sm[8];
    int b    = blockIdx.x;
    int lane = threadIdx.x & 31;
    int wave = threadIdx.x >> 5;
    float acc = 0.f;
    for (int n = threadIdx.x; n < N_; n += 256) {
        float v = qw[b * N_ + n];
        acc += v * v;
    }
    #pragma unroll
    for (int off = 16; off > 0; off >>= 1) acc += __shfl_down(acc, off, 32);
    if (lane == 0) sm[wave] = acc;
    __syncthreads();
    if (threadIdx.x == 0) {
        float s = 0.f;
        #pragma unroll
        for (int i = 0; i < 8; ++i) s += sm[i];
        invn[b] = 1.0f / fmaxf(sqrtf(s), EPS_);
    }
}

// ---------------------------------------------------------------------------
// K3: gpart[seg][b][d] = invn[b] * sum_{n in seg} qw[b,n] * x[b,n,d]
//     grid = (3, 8, 32), block = 256, fixed-order inner loop (deterministic)
// ---------------------------------------------------------------------------
__global__ void k_gpart(const float* __restrict__ x, const float* __restrict__ qw,
                        const float* __restrict__ invn, float* __restrict__ gpart) {
    const int SEG = N_ / 8;
    int d   = blockIdx.x * 256 + threadIdx.x;   // 0..767
    int seg = blockIdx.y;
    int b   = blockIdx.z;
    int n0  = seg * SEG;
    const float* xb = x + ((size_t)b * N_ + n0) * D_ + d;
    const float* qb = qw + b * N_ + n0;
    float acc = 0.f;
    for (int n = 0; n < SEG; ++n) acc += qb[n] * xb[(size_t)n * D_];
    gpart[((size_t)seg * B_ + b) * D_ + d] = acc * invn[b];
}

// ---------------------------------------------------------------------------
// K4: G[b][d] = sum over 8 segment partials
// ---------------------------------------------------------------------------
__global__ void k_greduce(const float* __restrict__ gpart, float* __restrict__ G) {
    int idx = blockIdx.x * 256 + threadIdx.x;   // 0 .. B*D-1
    float acc = 0.f;
    #pragma unroll
    for (int s = 0; s < 8; ++s) acc += gpart[(size_t)s * (B_ * D_) + idx];
    G[idx] = acc;
}

// ---------------------------------------------------------------------------
// K5: H = G @ proj_w^T  via V_WMMA_F32_16X16X4_F32 (full fp32 precision).
//     H[b][o] = sum_d G[b][d] * proj_w[o][d]
//     One wave32 per 16x16 output tile; grid = (48 ntiles, 2 mtiles).
//     fp32 A (16x4) fragment layout (ISA 7.12.2): lanes 0-15 hold K=0(v0),K=1(v1);
//     lanes 16-31 hold K=2(v0),K=3(v1). B (4x16) mirrors with N across lanes.
// ---------------------------------------------------------------------------
__global__ void k_wmma_proj(const float* __restrict__ G, const float* __restrict__ pw,
                            float* __restrict__ H) {
    int nt   = blockIdx.x;                // output-column tile (dout)
    int mt   = blockIdx.y;                // batch-row tile
    int lane = threadIdx.x;
    int half = lane >> 4;                 // 0: K0/K1, 1: K2/K3
    int l16  = lane & 15;
    int koff = half * 2;

    const float* Ar = G  + (size_t)(mt * 16 + l16) * D_;   // A row = batch index
    const float* Br = pw + (size_t)(nt * 16 + l16) * D_;   // B col = dout row of proj_w

    v8f c = {};
    for (int k = 0; k < D_; k += 4) {
        v2f a, b;
        a.x = Ar[k + koff];     a.y = Ar[k + koff + 1];
        b.x = Br[k + koff];     b.y = Br[k + koff + 1];
        c = __builtin_amdgcn_wmma_f32_16x16x4_f32(
                /*neg_a=*/false, a, /*neg_b=*/false, b,
                /*c_mod=*/(short)0, c, /*reuse_a=*/false, /*reuse_b=*/false);
    }
    // D layout: VGPR v -> row M = v + 8*half, col N = l16
    int rbase = mt * 16 + 8 * half;
    #pragma unroll
    for (int v = 0; v < 8; ++v)
        H[(size_t)(rbase + v) * D_ + nt * 16 + l16] = c[v];
}

// ---------------------------------------------------------------------------
// K6: ks[b*M+m] = sum_k key[b,m,k]   (one wave32 per row)
// ---------------------------------------------------------------------------
__global__ void k_ks(const float* __restrict__ key, float* __restrict__ ks) {
    int wave = threadIdx.x >> 5;
    int lane = threadIdx.x & 31;
    int row  = blockIdx.x * 8 + wave;           // 0 .. B*M-1
    const float* kr = key + (size_t)row * K2_;
    float acc = 0.f;
    #pragma unroll 4
    for (int i = lane; i < K2_; i += 32) acc += kr[i];
    #pragma unroll
    for (int off = 16; off > 0; off >>= 1) acc += __shfl_down(acc, off, 32);
    if (lane == 0) ks[row] = acc;
}

// ---------------------------------------------------------------------------
// K7: out[b,m,d] = ks[b,m] * H[b,d] + bias[d]  (float4 vectorized 100MB store)
// ---------------------------------------------------------------------------
__global__ void k_out(const float* __restrict__ ks, const float* __restrict__ H,
                      const float* __restrict__ bias, float* __restrict__ out) {
    size_t idx4 = (size_t)blockIdx.x * 256 + threadIdx.x;
    size_t f    = idx4 * 4;
    int    d    = (int)(f % D_);
    size_t bm   = f / D_;
    int    b    = (int)(bm / M_);
    float  s    = ks[bm];
    float4 h  = *(const float4*)(H + (size_t)b * D_ + d);
    float4 bb = *(const float4*)(bias + d);
    float4 o;
    o.x = s * h.x + bb.x;
    o.y = s * h.y + bb.y;
    o.z = s * h.z + bb.z;
    o.w = s * h.w + bb.w;
    *(float4*)(out + f) = o;
}

// ---------------------------------------------------------------------------
extern "C" void kernel_launch(void* const* d_in, const int* in_sizes, int n_in,
                              void* d_out, int out_size, void* d_ws, size_t ws_size,
                              hipStream_t stream) {
    const float* x      = (const float*)d_in[0];   // [B,N,D]
    const float* key    = (const float*)d_in[1];   // [B,M,K2]
    const float* w_g    = (const float*)d_in[2];   // [D,1]
    const float* proj_w = (const float*)d_in[3];   // [D,D]
    const float* proj_b = (const float*)d_in[4];   // [D]
    float* out = (float*)d_out;

    // Workspace layout (floats): ~1.25 MB total
    float* w     = (float*)d_ws;
    float* qw    = w;                    // B*N            = 32768
    float* invn  = w + 32768;            // B              = 32
    float* gpart = w + 32800;            // 8*B*D          = 196608
    float* G     = w + 229408;           // B*D            = 24576
    float* H     = w + 253984;           // B*D            = 24576
    float* ks    = w + 278560;           // B*M            = 32768

    // 1) qw = SCALE * x @ w_g
    k_qw<<<(B_ * N_) / 8, 256, 0, stream>>>(x, w_g, qw);
    // 2) per-batch inverse L2 norm
    k_norm<<<B_, 256, 0, stream>>>(qw, invn);
    // 3) segmented weighted column sums
    k_gpart<<<dim3(3, 8, B_), 256, 0, stream>>>(x, qw, invn, gpart);
    // 4) fold partials -> G
    k_greduce<<<(B_ * D_) / 256, 256, 0, stream>>>(gpart, G);
    // 5) H = G @ proj_w^T via fp32 WMMA
    k_wmma_proj<<<dim3(D_ / 16, 2), 32, 0, stream>>>(G, proj_w, H);
    // 6) ks = key row sums
    k_ks<<<(B_ * M_) / 8, 256, 0, stream>>>(key, ks);
    // 7) out = ks (x) H + bias   (rank-1 expansion + bias)
    k_out<<<(size_t)(B_ * M_ * D_) / 4 / 256, 256, 0, stream>>>(ks, H, proj_b, out);
}